// PullDown_23562190586021
// MI455X (gfx1250) — compile-verified
//
#include <hip/hip_runtime.h>

typedef __attribute__((ext_vector_type(2))) float v2f;
typedef __attribute__((ext_vector_type(8))) float v8f;

#define F_DIM 64
#define K_NB  32

// ---------------- Phase 0: zero the down_f accumulator in d_ws ----------------
__global__ void __launch_bounds__(256) pd_zero(float* __restrict__ p, int n) {
    int t = blockIdx.x * blockDim.x + threadIdx.x;
    if (t < n) p[t] = 0.0f;
}

// ---------------- Phase 1: scatter-add features into down_f ----------------
// One thread per (i, f) element; native f32 global atomics.
__global__ void __launch_bounds__(256) pd_scatter(const float* __restrict__ feat,
                                                  const int*   __restrict__ sel,
                                                  float*       __restrict__ down_f,
                                                  int total) {
    int t = blockIdx.x * blockDim.x + threadIdx.x;
    if (t >= total) return;
    int i = t >> 6;   // node index (F_DIM == 64)
    int f = t & 63;
    int d = sel[i];
    unsafeAtomicAdd(down_f + ((long)d << 6) + f, feat[t]);
}

// ---------------- Phase 2: weighted-mean gather via V_WMMA_F32_16X16X4_F32 ----
// One wave per output node m. out[m,:] = (1/K) * w(1x32) @ G(32x64).
// A (16x4 f32): weight values broadcast into all 16 rows  -> all D rows equal.
// B (4x16 f32): 4 gathered down_f rows for one 16-wide f-chunk.
// ISA layout: A VGPR0 = K0 (lanes 0-15) / K2 (lanes 16-31), VGPR1 = K1/K3;
//             B VGPR0 = row K0 / K2, VGPR1 = row K1 / K3 (N = lane & 15).
__global__ void __launch_bounds__(256) pd_gather_wmma(const float* __restrict__ down_f,
                                                      const float* __restrict__ wdown,
                                                      const int*   __restrict__ nidx,
                                                      float*       __restrict__ out,
                                                      int M) {
    const int lane = threadIdx.x & 31;
    const int wave = threadIdx.x >> 5;
    const int m    = blockIdx.x * 8 + wave;   // 8 waves / 256-thread block
    if (m >= M) return;                       // wave-uniform: EXEC stays all-1s

    // lane <-> neighbor k: load weight + index, apply mask (nidx<0 -> zero contrib)
    const long  kb      = (long)m * K_NB + lane;
    const int   idx_raw = nidx[kb];
    const float w_raw   = wdown[kb];
    const bool  ok      = idx_raw >= 0;
    const float w_eff   = ok ? w_raw  : 0.0f;
    const int   r_eff   = ok ? idx_raw : 0;

    const int col  = lane & 15;
    const int half = lane >> 4;

    v8f c0 = {0.f,0.f,0.f,0.f,0.f,0.f,0.f,0.f};
    v8f c1 = c0, c2 = c0, c3 = c0;

#pragma unroll
    for (int s = 0; s < 8; ++s) {
        const int k0   = s << 2;
        const int srcA = k0 + (half << 1);           // lanes 0-15: k0 ; 16-31: k0+2
        const float a0 = __shfl(w_eff, srcA,     32);
        const float a1 = __shfl(w_eff, srcA + 1, 32);
        const int   r0 = __shfl(r_eff, srcA,     32);
        const int   r1 = __shfl(r_eff, srcA + 1, 32);

        v2f A; A.x = a0; A.y = a1;

        const float* p0 = down_f + ((long)r0 << 6) + col;  // row for B VGPR0
        const float* p1 = down_f + ((long)r1 << 6) + col;  // row for B VGPR1

        v2f B0, B1, B2, B3;                       // 4 f-chunks of 16 columns
        B0.x = p0[0];   B0.y = p1[0];
        B1.x = p0[16];  B1.y = p1[16];
        B2.x = p0[32];  B2.y = p1[32];
        B3.x = p0[48];  B3.y = p1[48];

        c0 = __builtin_amdgcn_wmma_f32_16x16x4_f32(false, A, false, B0, (short)0, c0, false, false);
        c1 = __builtin_amdgcn_wmma_f32_16x16x4_f32(false, A, false, B1, (short)0, c1, false, false);
        c2 = __builtin_amdgcn_wmma_f32_16x16x4_f32(false, A, false, B2, (short)0, c2, false, false);
        c3 = __builtin_amdgcn_wmma_f32_16x16x4_f32(false, A, false, B3, (short)0, c3, false, false);
    }

    // Every D row is identical; c[0] holds (row 0, N=col) for lanes 0-15 and
    // (row 8, N=col) for lanes 16-31 -> same value. Split f-chunks across halves.
    const float scale = 1.0f / (float)K_NB;       // mean divides by K always
    const float vA = half ? c2[0] : c0[0];
    const float vB = half ? c3[0] : c1[0];
    const long  ob = ((long)m << 6) + (half ? 32 : 0) + col;
    out[ob]      = vA * scale;
    out[ob + 16] = vB * scale;
}

extern "C" void kernel_launch(void* const* d_in, const int* in_sizes, int n_in,
                              void* d_out, int out_size, void* d_ws, size_t ws_size,
                              hipStream_t stream) {
    const float* feat  = (const float*)d_in[0];   // features      [N,64] f32
    const float* wdown = (const float*)d_in[1];   // weights_down  [M,32] f32
    const int*   sel   = (const int*)  d_in[2];   // sel_idx_up    [N]
    const int*   nidx  = (const int*)  d_in[3];   // nidx_down     [M,32]
    float*       out   = (float*)d_out;           // [M,64] f32

    const int N = in_sizes[2];
    const int M = in_sizes[3] / K_NB;

    float* down_f = (float*)d_ws;                 // M*64 f32 scratch (25.6 MB)

    const int dn = M * F_DIM;
    pd_zero<<<(dn + 255) / 256, 256, 0, stream>>>(down_f, dn);

    const int st = N * F_DIM;
    pd_scatter<<<(st + 255) / 256, 256, 0, stream>>>(feat, sel, down_f, st);

    pd_gather_wmma<<<(M + 7) / 8, 256, 0, stream>>>(down_f, wdown, nidx, out, M);
}